// _GINEStack_48455821033920
// MI455X (gfx1250) — compile-verified
//
#include <hip/hip_runtime.h>

// Problem sizes (compile-time constants from the reference)
#define NNODES 50000
#define NEDGES 800000
#define IN_F   64
#define HD     128
#define EDD    16
#define NLAYERS 3
#define LN_EPS_F 1e-5f

typedef __attribute__((ext_vector_type(2))) float        v2f;
typedef __attribute__((ext_vector_type(8))) float        v8f;
typedef __attribute__((ext_vector_type(4))) unsigned int v4u;
typedef __attribute__((ext_vector_type(8))) int          v8i;
typedef __attribute__((ext_vector_type(4))) int          v4i;

// D = A(16x4 f32) * B(4x16 f32) + C(16x16 f32), full-precision CDNA5 WMMA.
__device__ __forceinline__ v8f wmma_f32_4(v2f a, v2f b, v8f c) {
    return __builtin_amdgcn_wmma_f32_16x16x4_f32(false, a, false, b, (short)0, c,
                                                 false, false);
}

// ---------------------------------------------------------------------------
// TDM: DMA a row-major [rows x cols] fp32 tile from global into LDS.
// D# packing per CDNA5 ISA ch.8 (group0: count/lds/global/type=2,
// group1: data_size=4B, dims, tile dims, stride). Issue from ONE wave only.
// This toolchain's builtin is the 6-arg form:
//   (v4u g0, v8i g1, v4i g2, v4i g3, v8i g4, i32 cpol)
// ---------------------------------------------------------------------------
__device__ __forceinline__ void tdm_load_tile_f32(const float* gptr, void* lds,
                                                  unsigned rows, unsigned cols,
                                                  unsigned row_stride_elems) {
#if __has_builtin(__builtin_amdgcn_tensor_load_to_lds)
    const unsigned long long ga = (unsigned long long)(uintptr_t)gptr;
    const unsigned lds_addr = (unsigned)(uintptr_t)lds;   // low 32b = LDS offset
    v4u g0;
    g0[0] = 1u;                                   // count=1, user descriptor
    g0[1] = lds_addr;                             // lds_addr
    g0[2] = (unsigned)ga;                         // global_addr[31:0]
    g0[3] = ((unsigned)(ga >> 32) & 0x01FFFFFFu)  // global_addr[56:32]
            | (2u << 30);                         // type=2 ("image")
    v8i g1;
    g1[0] = (int)(2u << 16);                      // data_size=2 -> 4 bytes
    g1[1] = (int)((cols & 0xFFFFu) << 16);        // tensor_dim0[15:0] @ [63:48]
    g1[2] = (int)(((cols >> 16) & 0xFFFFu) | ((rows & 0xFFFFu) << 16));
    g1[3] = (int)(((rows >> 16) & 0xFFFFu) | ((cols & 0xFFFFu) << 16)); // tile_dim0
    g1[4] = (int)(rows & 0xFFFFu);                // tile_dim1 (tile_dim2=0)
    g1[5] = (int)row_stride_elems;                // tensor_dim0_stride[31:0]
    g1[6] = 0;                                    // stride0[47:32]|stride1[15:0]
    g1[7] = 0;
    v4i zz4 = {0, 0, 0, 0};
    v8i zz8 = {0, 0, 0, 0, 0, 0, 0, 0};
    __builtin_amdgcn_tensor_load_to_lds(g0, g1, zz4, zz4, zz8, 0);
    __builtin_amdgcn_s_wait_tensorcnt(0);
#else
    // Fallback: cooperative vector loads by the issuing wave.
    float* d = (float*)lds;
    const int lane = threadIdx.x & 31;
    for (unsigned r = 0; r < rows; ++r)
        for (unsigned c = lane; c < cols; c += 32)
            d[r * cols + c] = gptr[(size_t)r * row_stride_elems + c];
#endif
}

// ---------------------------------------------------------------------------
// GEMM: C[M,Nc] = act(A[M,K] @ B[K,Nc] + bias[Nc]), row-major.
// Block = 256 threads = 8 waves; block tile = 64 rows x 128 cols.
// Each wave owns one 16-col strip and 4 row-tiles (4 accumulators), so every
// B fragment is reused by 4 WMMAs (4 independent WMMAs per K-step hide the
// XDL pipeline latency). Tail rows: clamped loads, guarded stores.
// grid = (ceil(M/64), Nc/128).
// ---------------------------------------------------------------------------
__global__ void gemm_bias_kernel(const float* __restrict__ A,
                                 const float* __restrict__ B,
                                 const float* __restrict__ bias,
                                 float* __restrict__ C,
                                 int M, int K, int Nc, int do_relu) {
    const int wave   = threadIdx.x >> 5;
    const int lane   = threadIdx.x & 31;
    const int lo     = lane & 15;
    const int hi     = lane >> 4;
    const int mBase0 = blockIdx.x << 6;                    // 64 rows / block
    const int nBase  = (blockIdx.y << 7) + (wave << 4);

    const float* aptr[4];
    #pragma unroll
    for (int t = 0; t < 4; ++t) {
        int row = mBase0 + (t << 4) + lo;
        if (row > M - 1) row = M - 1;                      // clamp tail reads
        aptr[t] = A + (size_t)row * K + (hi << 1);
    }
    const float* bcol = B + (size_t)(hi << 1) * Nc + nBase + lo;

    v8f acc0 = {}, acc1 = {}, acc2 = {}, acc3 = {};
    for (int k0 = 0; k0 < K; k0 += 4) {
        __builtin_prefetch(aptr[0] + k0 + 32, 0, 0);       // global_prefetch_b8
        v2f b;
        b.x = bcol[(size_t)k0 * Nc];
        b.y = bcol[(size_t)(k0 + 1) * Nc];
        v2f a0, a1, a2, a3;
        const float2 f0 = *(const float2*)(aptr[0] + k0);
        const float2 f1 = *(const float2*)(aptr[1] + k0);
        const float2 f2 = *(const float2*)(aptr[2] + k0);
        const float2 f3 = *(const float2*)(aptr[3] + k0);
        a0.x = f0.x; a0.y = f0.y;
        a1.x = f1.x; a1.y = f1.y;
        a2.x = f2.x; a2.y = f2.y;
        a3.x = f3.x; a3.y = f3.y;
        acc0 = wmma_f32_4(a0, b, acc0);
        acc1 = wmma_f32_4(a1, b, acc1);
        acc2 = wmma_f32_4(a2, b, acc2);
        acc3 = wmma_f32_4(a3, b, acc3);
    }

    const float bs = bias[nBase + lo];
    v8f* accs[4] = {&acc0, &acc1, &acc2, &acc3};
    #pragma unroll
    for (int t = 0; t < 4; ++t) {
        #pragma unroll
        for (int r = 0; r < 8; ++r) {
            const int orow = mBase0 + (t << 4) + r + (hi << 3);
            if (orow < M) {
                float v = (*accs[t])[r] + bs;
                if (do_relu) v = fmaxf(v, 0.0f);
                C[(size_t)orow * Nc + nBase + lo] = v;
            }
        }
    }
}

// ---------------------------------------------------------------------------
// Fused edge phase, 64 edges per block:
//   e = ea @ W(16x128) + bias ; msg = relu(e + h[src]) ; atomicAdd aggr[dst].
// W (8 KB) is DMA'd into LDS once per block via the Tensor Data Mover; the
// 8 waves then read their B fragments from LDS and keep them in registers
// across 4 edge sub-tiles (16 WMMAs per wave per block).
// grid = E/64, block = 256.
// ---------------------------------------------------------------------------
__global__ void edge_msg_kernel(const float* __restrict__ ea,
                                const float* __restrict__ W,
                                const float* __restrict__ bias,
                                const float* __restrict__ h,
                                const long long* __restrict__ src,
                                const long long* __restrict__ dst,
                                float* __restrict__ aggr) {
    __shared__ float s_W[EDD * HD];                  // 16x128 weights, 8 KB
    __shared__ int   s_src[64];
    __shared__ int   s_dst[64];

    const int eBase = blockIdx.x << 6;               // 64 edges / block
    if (threadIdx.x < 64) {
        s_src[threadIdx.x] = (int)src[eBase + threadIdx.x];
        s_dst[threadIdx.x] = (int)dst[eBase + threadIdx.x];
    }
    const int wave = threadIdx.x >> 5;
    if (wave == 0) {                                 // one wave issues the DMA
        tdm_load_tile_f32(W, s_W, EDD, HD, HD);
    }
    __syncthreads();

    const int lane = threadIdx.x & 31;
    const int lo   = lane & 15;
    const int hi   = lane >> 4;
    const int col  = (wave << 4) + lo;

    // B fragments from LDS, reused for all 4 edge sub-tiles.
    v2f wb[4];
    #pragma unroll
    for (int k = 0; k < 4; ++k) {
        const int kr = (k << 2) + (hi << 1);
        wb[k].x = s_W[kr * HD + col];
        wb[k].y = s_W[(kr + 1) * HD + col];
    }
    const float bs = bias[col];

    #pragma unroll
    for (int t = 0; t < 4; ++t) {
        const int rowBase = eBase + (t << 4);
        const float* arow = ea + (size_t)(rowBase + lo) * EDD + (hi << 1);
        v8f acc = {};
        #pragma unroll
        for (int k = 0; k < 4; ++k) {
            const float2 f = *(const float2*)(arow + (k << 2));
            v2f a; a.x = f.x; a.y = f.y;
            acc = wmma_f32_4(a, wb[k], acc);
        }
        #pragma unroll
        for (int r = 0; r < 8; ++r) {
            const int eidx = (t << 4) + r + (hi << 3);   // edge within block
            float msg = acc[r] + bs + h[(size_t)s_src[eidx] * HD + col];
            msg = fmaxf(msg, 0.0f);
            atomicAdd(&aggr[(size_t)s_dst[eidx] * HD + col], msg);
        }
    }
}

// ---------------------------------------------------------------------------
__global__ void fill_zero_kernel(float* __restrict__ p, int n) {
    const int i = blockIdx.x * blockDim.x + threadIdx.x;
    if (i < n) p[i] = 0.0f;
}

// z = (1 + eps[l]) * h + aggr
__global__ void combine_kernel(const float* __restrict__ h,
                               const float* __restrict__ aggr,
                               const float* __restrict__ eps, int l,
                               float* __restrict__ z, int n) {
    const int i = blockIdx.x * blockDim.x + threadIdx.x;
    if (i < n) z[i] = (1.0f + eps[l]) * h[i] + aggr[i];
}

// out = relu(layernorm(z) * g + b); one wave32 per 128-wide row.
__global__ void ln_relu_kernel(const float* __restrict__ z,
                               const float* __restrict__ g,
                               const float* __restrict__ b,
                               float* __restrict__ out) {
    const int wave = threadIdx.x >> 5;
    const int lane = threadIdx.x & 31;
    const int row  = (blockIdx.x << 3) + wave;

    const float4 v = ((const float4*)(z + (size_t)row * HD))[lane];
    float s  = v.x + v.y + v.z + v.w;
    float sq = v.x * v.x + v.y * v.y + v.z * v.z + v.w * v.w;
    #pragma unroll
    for (int off = 16; off > 0; off >>= 1) {
        s  += __shfl_xor(s, off, 32);
        sq += __shfl_xor(sq, off, 32);
    }
    const float mean = s * (1.0f / HD);
    const float var  = sq * (1.0f / HD) - mean * mean;
    const float rs   = rsqrtf(var + LN_EPS_F);

    const int c0 = lane << 2;
    float4 o;
    o.x = fmaxf((v.x - mean) * rs * g[c0 + 0] + b[c0 + 0], 0.0f);
    o.y = fmaxf((v.y - mean) * rs * g[c0 + 1] + b[c0 + 1], 0.0f);
    o.z = fmaxf((v.z - mean) * rs * g[c0 + 2] + b[c0 + 2], 0.0f);
    o.w = fmaxf((v.w - mean) * rs * g[c0 + 3] + b[c0 + 3], 0.0f);
    ((float4*)(out + (size_t)row * HD))[lane] = o;
}

// ---------------------------------------------------------------------------
extern "C" void kernel_launch(void* const* d_in, const int* in_sizes, int n_in,
                              void* d_out, int out_size, void* d_ws, size_t ws_size,
                              hipStream_t stream) {
    const float*     x       = (const float*)d_in[0];
    const long long* ei      = (const long long*)d_in[1];   // int64 [2, E]
    const float*     ea      = (const float*)d_in[2];
    const float*     proj_w  = (const float*)d_in[3];
    const float*     proj_b  = (const float*)d_in[4];
    const float*     lin_e_w = (const float*)d_in[5];
    const float*     lin_e_b = (const float*)d_in[6];
    const float*     eps     = (const float*)d_in[7];
    const float*     mlp_w1  = (const float*)d_in[8];
    const float*     mlp_b1  = (const float*)d_in[9];
    const float*     mlp_w2  = (const float*)d_in[10];
    const float*     mlp_b2  = (const float*)d_in[11];
    const float*     ln_g    = (const float*)d_in[12];
    const float*     ln_b    = (const float*)d_in[13];
    float*           out     = (float*)d_out;

    // Workspace layout (floats): h | z | aggr | t(2H)
    float* h    = (float*)d_ws;
    float* zbuf = h    + (size_t)NNODES * HD;
    float* aggr = zbuf + (size_t)NNODES * HD;
    float* tbuf = aggr + (size_t)NNODES * HD;

    const long long* src = ei;             // ei[0]
    const long long* dst = ei + NEDGES;    // ei[1]

    const dim3 blk(256);
    const int nh = NNODES * HD;
    const int gx = (NNODES + 63) / 64;     // 64-row GEMM blocks

    // h = x @ proj_w + proj_b
    gemm_bias_kernel<<<dim3(gx, HD / 128), blk, 0, stream>>>(
        x, proj_w, proj_b, h, NNODES, IN_F, HD, 0);

    for (int l = 0; l < NLAYERS; ++l) {
        fill_zero_kernel<<<(nh + 255) / 256, blk, 0, stream>>>(aggr, nh);

        edge_msg_kernel<<<NEDGES / 64, blk, 0, stream>>>(
            ea, lin_e_w + (size_t)l * EDD * HD, lin_e_b + (size_t)l * HD,
            h, src, dst, aggr);

        combine_kernel<<<(nh + 255) / 256, blk, 0, stream>>>(
            h, aggr, eps, l, zbuf, nh);

        // t = relu(z @ w1 + b1)   [N,128] @ [128,256]
        gemm_bias_kernel<<<dim3(gx, (2 * HD) / 128), blk, 0, stream>>>(
            zbuf, mlp_w1 + (size_t)l * HD * 2 * HD, mlp_b1 + (size_t)l * 2 * HD,
            tbuf, NNODES, HD, 2 * HD, 1);

        // z = t @ w2 + b2         [N,256] @ [256,128]  (overwrites zbuf)
        gemm_bias_kernel<<<dim3(gx, HD / 128), blk, 0, stream>>>(
            tbuf, mlp_w2 + (size_t)l * 2 * HD * HD, mlp_b2 + (size_t)l * HD,
            zbuf, NNODES, 2 * HD, HD, 0);

        float* dest = (l == NLAYERS - 1) ? out : h;
        ln_relu_kernel<<<NNODES / 8, blk, 0, stream>>>(
            zbuf, ln_g + (size_t)l * HD, ln_b + (size_t)l * HD, dest);
    }
}